// MeshConv_72937134621192
// MI455X (gfx1250) — compile-verified
//
#include <hip/hip_runtime.h>

typedef __attribute__((ext_vector_type(2))) float v2f;
typedef __attribute__((ext_vector_type(4))) float v4f;
typedef __attribute__((ext_vector_type(8))) float v8f;

#define N_B   4
#define CIN   16
#define COUT  16
#define F_IN  500000
#define KW    5
#define FOUT  (F_IN - KW + 1)   // 499996
#define FT    64                 // output f positions per block (four 16-wide WMMA tiles)
#define FL    (FT + KW - 1)      // 68 staged g positions per (ci,h)
#define GRP_PER_CI (FL / 4)      // 17 groups of 4 f-positions
#define NGRP  (CIN * GRP_PER_CI) // 272
#define NTHREADS 160             // 5 waves: wave w handles h-plane w

__global__ __launch_bounds__(NTHREADS)
void meshconv_wmma_f32(const float* __restrict__ x,
                       const float* __restrict__ W,
                       const float* __restrict__ bias,
                       float* __restrict__ y)
{
    // g staged as [ci][h][fl]; W staged row-major 16x80 (A[co][kk], kk=ci*5+k)
    __shared__ float g_lds[CIN * 5 * FL];   // 21760 B
    __shared__ float w_lds[COUT * 80];      //  5120 B

    const int tid = threadIdx.x;
    const int n   = blockIdx.y;
    const long f0 = (long)blockIdx.x * FT;

    // ---------------- Phase 1a: stage W via float4 copies ----------------
    for (int p = tid; p < (COUT * 80) / 4; p += NTHREADS)
        ((v4f*)w_lds)[p] = ((const v4f*)W)[p];

    // ---------------- Phase 1b: vector-load x, mesh transform, stage g ----------------
    // Each group = one (ci, 4 consecutive f) strip = 20 contiguous floats, 16B aligned.
    for (int p = tid; p < NGRP; p += NTHREADS) {
        const int ci  = p / GRP_PER_CI;
        const int fl0 = (p - ci * GRP_PER_CI) * 4;
        const long f  = f0 + fl0;
        const float* xp = x + (((long)(n * CIN + ci) * F_IN) + f) * KW;

        float v[20];
        if (f + 3 < F_IN) {
            #pragma unroll
            for (int q = 0; q < 5; ++q) {              // 5x global_load_b128
                v4f t = ((const v4f*)xp)[q];
                v[4*q+0] = t.x; v[4*q+1] = t.y; v[4*q+2] = t.z; v[4*q+3] = t.w;
            }
        } else {
            #pragma unroll
            for (int e = 0; e < 20; ++e)               // ragged tail (last f-tile only)
                v[e] = (f + (e / 5) < F_IN) ? xp[e] : 0.f;
        }

        // v[j*5+k] = x[ci][f+j][k]; build 4-wide g rows, one per h-plane
        v4f G0, G1, G2, G3, G4;
        #pragma unroll
        for (int j = 0; j < 4; ++j) {
            G0[j] = v[5*j + 0];
            G1[j] = v[5*j + 1] + v[5*j + 3];
            G2[j] = v[5*j + 2] + v[5*j + 4];
            G3[j] = fabsf(v[5*j + 1] - v[5*j + 3]);
            G4[j] = fabsf(v[5*j + 2] - v[5*j + 4]);
        }
        v4f* gp = (v4f*)(g_lds + ci * (5 * FL) + fl0); // 16B-aligned
        gp[0 * (FL/4)] = G0;                            // 5x ds_store_b128
        gp[1 * (FL/4)] = G1;
        gp[2 * (FL/4)] = G2;
        gp[3 * (FL/4)] = G3;
        gp[4 * (FL/4)] = G4;
    }
    __syncthreads();

    // ---------------- Phase 2: per-wave GEMM over one h-plane ----------------
    const int wave = tid >> 5;      // h-plane index 0..4
    const int lane = tid & 31;
    const int m    = lane & 15;     // A: M row (=co); B/C/D: N col (=f offset)
    const int hi   = lane >> 4;     // lane-half -> K sub-rows {0,1} vs {2,3}

    // A fragments: 16x4 f32 layout — lane holds M=m; VGPR0=K(2*hi), VGPR1=K(2*hi+1)
    v2f a[20];
    #pragma unroll
    for (int g = 0; g < 20; ++g) {
        const int kk = 4 * g + 2 * hi;
        a[g].x = w_lds[m * 80 + kk];
        a[g].y = w_lds[m * 80 + kk + 1];
    }

    v8f c0 = {}, c1 = {}, c2 = {}, c3 = {};  // f-cols f0 + {0,16,32,48} + m
    const float* gh = &g_lds[wave * FL];     // + ci*(5*FL) + pos

    #pragma unroll
    for (int g = 0; g < 20; ++g) {
        // B fragments (4x16, mirror of A layout): VGPR j holds row K=4g+2*hi+j, col N=m
        v2f b0, b1, b2, b3;
        {
            const int kk = 4 * g + 2 * hi;       // j = 0
            const int ci = kk / 5;
            const int k  = kk - 5 * ci;
            const float* bp = gh + ci * (5 * FL) + m + k;
            b0.x = bp[0]; b1.x = bp[16]; b2.x = bp[32]; b3.x = bp[48];
        }
        {
            const int kk = 4 * g + 2 * hi + 1;   // j = 1
            const int ci = kk / 5;
            const int k  = kk - 5 * ci;
            const float* bp = gh + ci * (5 * FL) + m + k;
            b0.y = bp[0]; b1.y = bp[16]; b2.y = bp[32]; b3.y = bp[48];
        }
        c0 = __builtin_amdgcn_wmma_f32_16x16x4_f32(false, a[g], false, b0,
                                                   (short)0, c0, false, false);
        c1 = __builtin_amdgcn_wmma_f32_16x16x4_f32(false, a[g], false, b1,
                                                   (short)0, c1, false, false);
        c2 = __builtin_amdgcn_wmma_f32_16x16x4_f32(false, a[g], false, b2,
                                                   (short)0, c2, false, false);
        c3 = __builtin_amdgcn_wmma_f32_16x16x4_f32(false, a[g], false, b3,
                                                   (short)0, c3, false, false);
    }

    // ---------------- Phase 3: bias + guarded stores ----------------
    #pragma unroll
    for (int r = 0; r < 8; ++r) {
        const int co = r + 8 * hi;              // C/D layout: VGPR r -> M = r + 8*hi
        const float bv = bias[co];
        const long rowBase = ((long)(n * COUT + co) * 5 + wave) * (long)FOUT;
        const long fA = f0 + m;
        if (fA      < FOUT) y[rowBase + fA]      = c0[r] + bv;
        if (fA + 16 < FOUT) y[rowBase + fA + 16] = c1[r] + bv;
        if (fA + 32 < FOUT) y[rowBase + fA + 32] = c2[r] + bv;
        if (fA + 48 < FOUT) y[rowBase + fA + 48] = c3[r] + bv;
    }
}

extern "C" void kernel_launch(void* const* d_in, const int* in_sizes, int n_in,
                              void* d_out, int out_size, void* d_ws, size_t ws_size,
                              hipStream_t stream)
{
    const float* x  = (const float*)d_in[0];
    const float* W  = (const float*)d_in[1];
    const float* b  = (const float*)d_in[2];
    float* y = (float*)d_out;

    dim3 grid((FOUT + FT - 1) / FT, N_B);   // (7813, 4)
    meshconv_wmma_f32<<<grid, NTHREADS, 0, stream>>>(x, W, b, y);
}